// CloudCastLoss_67473936220950
// MI455X (gfx1250) — compile-verified
//
#include <hip/hip_runtime.h>
#include <hip/hip_bf16.h>
#include <math.h>

typedef float v2f __attribute__((ext_vector_type(2)));
typedef float v8f __attribute__((ext_vector_type(8)));

#define HIST   1024
#define TPB    256
#define BN     8
#define NPIX   (768*768)          // 589824 per sample
#define GX     64                 // blocks per sample
#define V4PB   (NPIX/4/GX)        // 2304 float4 per block
#define V4PT   (V4PB/TPB)         // 9 float4 per thread
#define FMAXV  3.5f               // max negative focal = 0.25*(-log(1e-6)) ~ 3.45
#define EPSV   1e-6f
#define RATIO  10

#if defined(__HIP_DEVICE_COMPILE__)
#  if __has_builtin(__builtin_amdgcn_wmma_f32_16x16x4_f32)
#    define USE_WMMA_RED 1
#  endif
#endif

__global__ void zero_ws_kernel(unsigned* __restrict__ w, int n) {
    int i = blockIdx.x * blockDim.x + threadIdx.x;
    if (i < n) w[i] = 0u;
}

// Fused streaming pass: focal(+histogram of negatives), tversky sums, huber reg sums.
__global__ __launch_bounds__(TPB)
void main_pass(const float* __restrict__ prob, const float* __restrict__ lab,
               const float* __restrict__ rlog, const float* __restrict__ rsp,
               unsigned* __restrict__ hcnt, float* __restrict__ hsum,
               float* __restrict__ acc)
{
    __shared__ unsigned scnt[HIST];
    __shared__ float    ssum[HIST];
    __shared__ float    wp[16][8];   // rows: 7 quantities (rest zero), cols: 8 waves
    const int tid = threadIdx.x;
    const int b   = blockIdx.y;

    for (int i = tid; i < HIST; i += TPB) { scnt[i] = 0u; ssum[i] = 0.0f; }
    if (tid < 128) ((float*)wp)[tid] = 0.0f;
    __syncthreads();

    const size_t base = (size_t)b * NPIX;
    const float4* p4 = (const float4*)(prob + base) + (size_t)blockIdx.x * V4PB;
    const float4* t4 = (const float4*)(lab  + base) + (size_t)blockIdx.x * V4PB;
    const float4* r4 = (const float4*)(rlog + base) + (size_t)blockIdx.x * V4PB;
    const float4* q4 = (const float4*)(rsp  + base) + (size_t)blockIdx.x * V4PB;

    float posS = 0.f, npos = 0.f, tpv = 0.f, spv = 0.f, stv = 0.f, hubS = 0.f, wS = 0.f;

    for (int i = 0; i < V4PT; i++) {
        const int idx = tid + i * TPB;
        __builtin_prefetch(p4 + idx + TPB, 0, 3);   // global_prefetch_b8 (speculative)
        __builtin_prefetch(t4 + idx + TPB, 0, 3);
        __builtin_prefetch(r4 + idx + TPB, 0, 3);
        __builtin_prefetch(q4 + idx + TPB, 0, 3);
        const float4 P = p4[idx], T = t4[idx], R = r4[idx], Q = q4[idx];
#pragma unroll
        for (int e = 0; e < 4; e++) {
            const float pr = (&P.x)[e], t = (&T.x)[e], rl = (&R.x)[e], rs = (&Q.x)[e];
            const float p  = fminf(fmaxf(pr, EPSV), 1.0f - EPSV);
            const float lg = logf(p), l1 = log1pf(-p);
            const float bce = -(2.0f * t * lg + (1.0f - t) * l1);  // POS_W=2
            const bool  pos = (t >= 0.5f);
            const float pt  = pos ? p : 1.0f - p;
            const float at  = pos ? 0.75f : 0.25f;
            const float om  = 1.0f - pt;
            const float focal = at * om * om * bce;
            if (pos) { posS += focal; npos += 1.0f; }
            else {
                int bin = (int)(focal * ((float)HIST / FMAXV));
                bin = bin < 0 ? 0 : (bin > HIST - 1 ? HIST - 1 : bin);
                atomicAdd(&scnt[bin], 1u);
                atomicAdd(&ssum[bin], focal);
            }
            tpv += p * t; spv += p; stv += t;
            const float wgt = ((pr > 0.1f) || (rs > 1.0f)) ? (rs >= 50.0f ? 4.0f : 1.0f) : 0.0f;
            if (wgt != 0.0f) {
                const float er = (rl - log1pf(fmaxf(rs, 0.0f))) * 2.0f;  // /RAIN_LOG_STD
                const float ae = fabsf(er);
                const float hb = (ae <= 1.0f) ? 0.5f * er * er : ae - 0.5f;
                hubS += hb * wgt; wS += wgt;
            }
        }
    }

    // intra-wave (wave32) reduction
#pragma unroll
    for (int o = 16; o; o >>= 1) {
        posS += __shfl_down(posS, o); npos += __shfl_down(npos, o);
        tpv  += __shfl_down(tpv,  o); spv  += __shfl_down(spv,  o);
        stv  += __shfl_down(stv,  o); hubS += __shfl_down(hubS, o);
        wS   += __shfl_down(wS,   o);
    }
    const int lane = tid & 31, wid = tid >> 5;
    if (lane == 0) {
        wp[0][wid] = posS; wp[1][wid] = npos; wp[2][wid] = tpv; wp[3][wid] = spv;
        wp[4][wid] = stv;  wp[5][wid] = hubS; wp[6][wid] = wS;
    }
    __syncthreads();

    // flush LDS histogram to per-sample global histogram
    for (int i = tid; i < HIST; i += TPB) {
        const unsigned c = scnt[i];
        if (c) {
            atomicAdd(&hcnt[(size_t)b * HIST + i], c);
            atomicAdd(&hsum[(size_t)b * HIST + i], ssum[i]);
        }
    }

    // cross-wave reduction of the 7 block accumulators.
    if (tid < 32) {   // wave 0: EXEC all-ones inside this branch
#ifdef USE_WMMA_RED
        // A (16x4 f32): row m = quantity m, K = wave partials; B = ones(4x16).
        // D(m,n) = sum_k A(m,k): exact f32 rowsum via tensor core, accumulated over 2 K-chunks.
        const int m = tid & 15, half = tid >> 4;
        v2f ones; ones[0] = 1.0f; ones[1] = 1.0f;
        v8f d = {};
#pragma unroll
        for (int j = 0; j < 2; j++) {
            v2f a;
            a[0] = wp[m][j * 4 + half * 2 + 0];
            a[1] = wp[m][j * 4 + half * 2 + 1];
            d = __builtin_amdgcn_wmma_f32_16x16x4_f32(false, a, false, ones,
                                                      (short)0, d, false, false);
        }
        if (tid == 0) {   // lane 0 holds D(m,0) for m=0..7 in d[0..7]
            atomicAdd(&acc[b * 8 + 0], d[0]);
            atomicAdd(&acc[b * 8 + 1], d[1]);
            atomicAdd(&acc[b * 8 + 2], d[2]);
            atomicAdd(&acc[b * 8 + 3], d[3]);
            atomicAdd(&acc[b * 8 + 4], d[4]);
            atomicAdd(&acc[b * 8 + 5], d[5]);
            atomicAdd(&acc[b * 8 + 6], d[6]);
        }
#else
        if (tid == 0) {
            for (int m2 = 0; m2 < 7; m2++) {
                float s = 0.0f;
                for (int w2 = 0; w2 < 8; w2++) s += wp[m2][w2];
                atomicAdd(&acc[b * 8 + m2], s);
            }
        }
#endif
    }
}

// One block. Wave b scans sample b's histogram (two-level, top-down) for the top-k
// negative-focal sum, then thread 0 combines fl/tv/reg/aux into the 5 outputs.
__global__ __launch_bounds__(TPB)
void finalize_pass(const unsigned* __restrict__ hcnt, const float* __restrict__ hsum,
                   const float* __restrict__ acc,
                   const float* __restrict__ pred_phys, const float* __restrict__ phys_tgt,
                   const float* __restrict__ mu, const float* __restrict__ sd,
                   float* __restrict__ out)
{
    __shared__ float    red[128];
    __shared__ unsigned chC[BN][32];
    __shared__ float    chS[BN][32];
    __shared__ float    perS[BN], tvS[BN];
    __shared__ float    s_aux;
    const int tid = threadIdx.x;

    // aux = mean((pred_phys - nan_to_num((tgt-mu)/(std+1e-6)))^2) over 8x16
    if (tid < 128) {
        const int j = tid & 15;
        float t = (phys_tgt[tid] - mu[j]) / (sd[j] + 1e-6f);
        if (isnan(t)) t = 0.0f;
        const float dd = pred_phys[tid] - t;
        red[tid] = dd * dd;
    }
    __syncthreads();
    if (tid == 0) { float s = 0.f; for (int i = 0; i < 128; i++) s += red[i]; s_aux = s / 128.0f; }

    const int b = tid >> 5, lane = tid & 31;
    // level-1: per-lane 32-bin chunk totals
    unsigned cc = 0; float cs = 0.f;
    for (int k = 0; k < 32; k++) {
        const int bin = lane * 32 + k;
        cc += hcnt[b * HIST + bin];
        cs += hsum[b * HIST + bin];
    }
    chC[b][lane] = cc; chS[b][lane] = cs;
    __syncthreads();

    if (lane == 0) {
        const float posS = acc[b * 8 + 0];
        const long long npos = (long long)(acc[b * 8 + 1] + 0.5f);
        const float tp = acc[b * 8 + 2], sp = acc[b * 8 + 3], st = acc[b * 8 + 4];
        const long long nneg = (long long)NPIX - npos;
        long long nh = npos * RATIO; if (nh > nneg) nh = nneg;

        unsigned long long totC = 0; float negS = 0.f;
        for (int c = 0; c < 32; c++) { totC += chC[b][c]; negS += chS[b][c]; }

        float topS = 0.0f;
        if (npos > 0 && nh > 0) {
            unsigned long long accC = 0; float accS = 0.f;
            for (int c = 31; c >= 0; c--) {
                const unsigned ccc = chC[b][c];
                if (accC + ccc >= (unsigned long long)nh) {
                    for (int k = 31; k >= 0; k--) {
                        const int bin = c * 32 + k;
                        const unsigned bc = hcnt[b * HIST + bin];
                        const float    bs = hsum[b * HIST + bin];
                        if (accC + bc >= (unsigned long long)nh) {
                            const unsigned long long need = (unsigned long long)nh - accC;
                            topS = accS + (bc ? ((float)need / (float)bc) * bs : 0.0f);
                            break;
                        }
                        accC += bc; accS += bs;
                    }
                    break;
                }
                accC += ccc; accS += chS[b][c];
            }
        }
        float per;
        if (npos > 0) {
            float den = (float)(npos + nh); if (den < 1.0f) den = 1.0f;
            per = (posS + topS) / den;
        } else {
            per = totC ? negS / (float)totC : 0.0f;   // ~mean of random negative subset
        }
        perS[b] = per;
        const float fp = sp - tp, fn = st - tp;
        tvS[b] = 1.0f - (tp + 1.0f) / (tp + 0.3f * fp + 0.7f * fn + 1.0f);
    }
    __syncthreads();

    if (tid == 0) {
        float fl = 0.f, tv = 0.f, hub = 0.f, wt = 0.f;
        for (int i = 0; i < BN; i++) {
            fl += perS[i]; tv += tvS[i];
            hub += acc[i * 8 + 5]; wt += acc[i * 8 + 6];
        }
        fl /= (float)BN; tv /= (float)BN;
        const float reg = hub / fmaxf(wt, 1.0f);
        const float aux = s_aux;
        out[0] = fl + 0.5f * tv + reg + 0.1f * aux;
        out[1] = fl; out[2] = tv; out[3] = reg; out[4] = aux;
    }
}

extern "C" void kernel_launch(void* const* d_in, const int* in_sizes, int n_in,
                              void* d_out, int out_size, void* d_ws, size_t ws_size,
                              hipStream_t stream)
{
    (void)in_sizes; (void)n_in; (void)out_size; (void)ws_size;
    const float* prob  = (const float*)d_in[0];   // (8,768,768)
    const float* rlog  = (const float*)d_in[1];   // (8,768,768)
    const float* pphys = (const float*)d_in[2];   // (8,16)
    const float* lab   = (const float*)d_in[3];   // (8,768,768)
    /* d_in[4] rain_max_true: unused by reference */
    const float* rsp   = (const float*)d_in[5];   // (8,768,768)
    const float* ptgt  = (const float*)d_in[6];   // (8,16)
    const float* mu    = (const float*)d_in[7];   // (16,)
    const float* sd    = (const float*)d_in[8];   // (16,)

    unsigned* hcnt = (unsigned*)d_ws;                       // 8*1024 u32
    float*    hsum = (float*)d_ws + BN * HIST;              // 8*1024 f32
    float*    acc  = (float*)d_ws + 2 * BN * HIST;          // 8*8 f32

    const int zeroN = 2 * BN * HIST + BN * 8;               // dwords to clear every call
    zero_ws_kernel<<<(zeroN + TPB - 1) / TPB, TPB, 0, stream>>>((unsigned*)d_ws, zeroN);
    main_pass<<<dim3(GX, BN), TPB, 0, stream>>>(prob, lab, rlog, rsp, hcnt, hsum, acc);
    finalize_pass<<<1, TPB, 0, stream>>>(hcnt, hsum, acc, pphys, ptgt, mu, sd, (float*)d_out);
}